// TestModel_48945447305605
// MI455X (gfx1250) — compile-verified
//
#include <hip/hip_runtime.h>
#include <math.h>
#include <stdint.h>

#define NN 262144
#define NE 2097152
#define EPSL 1e-5f

typedef __attribute__((ext_vector_type(2))) float v2f;
typedef __attribute__((ext_vector_type(8))) float v8f;

__device__ __forceinline__ void atomic_add_f32(float* p, float v) {
    __hip_atomic_fetch_add(p, v, __ATOMIC_RELAXED, __HIP_MEMORY_SCOPE_AGENT);
}

// ---------------- degree / normalization ----------------
__global__ void k_init_deg(float* deg) {
    int i = blockIdx.x * blockDim.x + threadIdx.x;
    if (i < NN) deg[i] = 1.0f;                       // +1 self-loop
}

__global__ void k_count_deg(float* deg, const int* __restrict__ dst) {
    int e = blockIdx.x * blockDim.x + threadIdx.x;
    if (e < NE) atomic_add_f32(&deg[dst[e]], 1.0f);
}

__global__ void k_rsqrt(float* dis) {
    int i = blockIdx.x * blockDim.x + threadIdx.x;
    if (i < NN) dis[i] = rsqrtf(dis[i]);
}

// ---------------- layer-1 linear: x[N,9] @ W[9,64], 4 outputs/thread -------
__global__ void k_linear9(const float* __restrict__ x, const float* __restrict__ W,
                          float* __restrict__ h) {
    int t = blockIdx.x * blockDim.x + threadIdx.x;   // NN*16 threads
    if (t >= NN * 16) return;
    int node = t >> 4, q = (t & 15) << 2;
    const float* xr = x + node * 9;
    float4 acc = make_float4(0.f, 0.f, 0.f, 0.f);
#pragma unroll
    for (int k = 0; k < 9; ++k) {
        float xv = xr[k];
        const float4 wv = *(const float4*)(W + k * 64 + q);
        acc.x = fmaf(xv, wv.x, acc.x);
        acc.y = fmaf(xv, wv.y, acc.y);
        acc.z = fmaf(xv, wv.z, acc.z);
        acc.w = fmaf(xv, wv.w, acc.w);
    }
    *(float4*)(h + (size_t)node * 64 + q) = acc;
}

// ---------------- fp32 WMMA linear: in[N,64] @ W[64,fout] ----------------
// W staged into LDS via async-to-LDS (ASYNCcnt path), B fragments from LDS.
// One wave per 16-row tile; K stepped by 4 using V_WMMA_F32_16X16X4_F32.
//   A (16x4 f32):  lane l -> M = l&15, K = (l>>4)*2 + {vgpr0,vgpr1}
//   B (4x16 f32):  lane l -> N = l&15, K = (l>>4)*2 + {vgpr0,vgpr1}
//   C/D (16x16):   vgpr v, lane l -> M = (l>>4)*8 + v, N = l&15
__global__ __launch_bounds__(128)
void k_linear_wmma(const float* __restrict__ in, const float* __restrict__ W,
                   float* __restrict__ out, int fout) {
    __shared__ float WS[64 * 64];                    // 16 KB max (64x64 f32)

    // ---- async copy of W (64*fout floats) into LDS, 16B per lane/iter ----
    {
        int nbytes = 64 * fout * 4;
        uint32_t lbase = (uint32_t)(uintptr_t)(&WS[0]);  // low 32b of flat = LDS offset
        uint64_t gbase = (uint64_t)(uintptr_t)W;
        for (int off = threadIdx.x * 16; off < nbytes; off += 128 * 16) {
            uint32_t laddr = lbase + (uint32_t)off;
            uint64_t gaddr = gbase + (uint64_t)off;
            asm volatile("global_load_async_to_lds_b128 %0, %1, off"
                         :: "v"(laddr), "v"(gaddr) : "memory");
        }
        asm volatile("s_wait_asynccnt 0x0" ::: "memory");
        __syncthreads();
    }

    int wave = blockIdx.x * 4 + (threadIdx.x >> 5);  // grid covers N/16 tiles exactly
    int lane = threadIdx.x & 31;
    int m0 = wave << 4;
    int row = lane & 15;
    int kh  = (lane >> 4) << 1;                      // 0 or 2
    const float* arow = in + (size_t)(m0 + row) * 64;

    v2f a[16];
#pragma unroll
    for (int ks = 0; ks < 16; ++ks)                  // 8B-aligned pair loads (b64)
        a[ks] = *(const v2f*)(arow + ks * 4 + kh);

    for (int n0 = 0; n0 < fout; n0 += 16) {
        v8f c = {};
#pragma unroll
        for (int ks = 0; ks < 16; ++ks) {
            int k = ks * 4 + kh;
            v2f b;
            b.x = WS[k * fout + n0 + row];
            b.y = WS[(k + 1) * fout + n0 + row];
            c = __builtin_amdgcn_wmma_f32_16x16x4_f32(
                    false, a[ks], false, b, (short)0, c, false, false);
        }
        int mbase = m0 + ((lane >> 4) << 3);
#pragma unroll
        for (int v = 0; v < 8; ++v)
            out[(size_t)(mbase + v) * fout + n0 + row] = c[v];
    }
}

// ---------------- self-loop + bias init, float4 (also zeroes LN cells) ----
__global__ void k_agg_init4(const float* __restrict__ h, const float* __restrict__ dis,
                            const float* __restrict__ bias, float* __restrict__ agg,
                            int fshift, float* red) {
    long t = (long)blockIdx.x * blockDim.x + threadIdx.x;
    if (t == 0) { red[0] = 0.f; red[1] = 0.f; }
    long total4 = ((long)NN << fshift) >> 2;
    if (t >= total4) return;
    long i = t << 2;
    int node = (int)(t >> (fshift - 2));
    int f = (int)(i & ((1 << fshift) - 1));
    float d = dis[node];
    float d2 = d * d;
    const float4 hv = *(const float4*)(h + i);
    const float4 bv = *(const float4*)(bias + f);
    float4 o;
    o.x = fmaf(hv.x, d2, bv.x);
    o.y = fmaf(hv.y, d2, bv.y);
    o.z = fmaf(hv.z, d2, bv.z);
    o.w = fmaf(hv.w, d2, bv.w);
    *(float4*)(agg + i) = o;
}

// ------- edge scatter, float4 gather + 4 native f32 atomics per thread ----
__global__ void k_agg_edges4(const float* __restrict__ h, const float* __restrict__ dis,
                             const int* __restrict__ src, const int* __restrict__ dst,
                             float* agg, int fshift) {
    int qs = fshift - 2;                              // log2(groups of 4 per edge)
    long t = (long)blockIdx.x * blockDim.x + threadIdx.x;
    long total = (long)NE << qs;
    if (t >= total) return;
    int e = (int)(t >> qs);                           // shared by 1<<qs consecutive lanes
    int q = (int)(t & ((1 << qs) - 1));
    int s = src[e], d = dst[e];
    float norm = dis[s] * dis[d];
    const float4 hv = *(const float4*)(h + ((long)s << fshift) + (q << 2));
    float* base = agg + ((long)d << fshift) + (q << 2);
    atomic_add_f32(base + 0, hv.x * norm);
    atomic_add_f32(base + 1, hv.y * norm);
    atomic_add_f32(base + 2, hv.z * norm);
    atomic_add_f32(base + 3, hv.w * norm);
}

// ---------------- global sum / sumsq for graph LayerNorm (float4) --------
__global__ void k_stats(const float4* __restrict__ x, long total4, float* red) {
    long t = (long)blockIdx.x * blockDim.x + threadIdx.x;
    long stride = (long)gridDim.x * blockDim.x;
    float s = 0.f, s2 = 0.f;
    for (long i = t; i < total4; i += stride) {
        const float4 v = x[i];
        s += v.x + v.y + v.z + v.w;
        s2 = fmaf(v.x, v.x, s2);
        s2 = fmaf(v.y, v.y, s2);
        s2 = fmaf(v.z, v.z, s2);
        s2 = fmaf(v.w, v.w, s2);
    }
#pragma unroll
    for (int o = 16; o > 0; o >>= 1) {               // wave32 reduction
        s  += __shfl_xor(s,  o, 32);
        s2 += __shfl_xor(s2, o, 32);
    }
    if ((threadIdx.x & 31) == 0) {
        atomic_add_f32(&red[0], s);
        atomic_add_f32(&red[1], s2);
    }
}

// ---------------- graph LN + ELU, float4 in place ----------------
__device__ __forceinline__ float elu1(float v) { return v > 0.f ? v : expm1f(v); }

__global__ void k_ln_elu4(float* x, const float* __restrict__ red,
                          const float* __restrict__ g, const float* __restrict__ beta,
                          int fshift) {
    long t = (long)blockIdx.x * blockDim.x + threadIdx.x;
    long total4 = ((long)NN << fshift) >> 2;
    if (t >= total4) return;
    long i = t << 2;
    float inv  = 1.0f / (float)((long)NN << fshift);
    float mean = red[0] * inv;
    float var  = fmaxf(red[1] * inv - mean * mean, 0.f);
    float istd = 1.0f / (sqrtf(var) + EPSL);
    int f = (int)(i & ((1 << fshift) - 1));
    const float4 v  = *(const float4*)(x + i);
    const float4 gv = *(const float4*)(g + f);
    const float4 bv = *(const float4*)(beta + f);
    float4 o;
    o.x = elu1(fmaf((v.x - mean) * istd, gv.x, bv.x));
    o.y = elu1(fmaf((v.y - mean) * istd, gv.y, bv.y));
    o.z = elu1(fmaf((v.z - mean) * istd, gv.z, bv.z));
    o.w = elu1(fmaf((v.w - mean) * istd, gv.w, bv.w));
    *(float4*)(x + i) = o;
}

__global__ void k_elu4(float* x, long total4) {
    long t = (long)blockIdx.x * blockDim.x + threadIdx.x;
    if (t >= total4) return;
    long i = t << 2;
    const float4 v = *(const float4*)(x + i);
    float4 o = make_float4(elu1(v.x), elu1(v.y), elu1(v.z), elu1(v.w));
    *(float4*)(x + i) = o;
}

// ---------------- layer-5: 16 -> 1, scalar feature ----------------
__global__ void k_linear16(const float* __restrict__ in, const float* __restrict__ W,
                           float* __restrict__ h) {
    int i = blockIdx.x * blockDim.x + threadIdx.x;
    if (i >= NN) return;
    const float4* r = (const float4*)(in + (size_t)i * 16);
    const float4* w = (const float4*)W;
    float acc = 0.f;
#pragma unroll
    for (int k = 0; k < 4; ++k) {
        const float4 rv = r[k], wv = w[k];
        acc = fmaf(rv.x, wv.x, acc);
        acc = fmaf(rv.y, wv.y, acc);
        acc = fmaf(rv.z, wv.z, acc);
        acc = fmaf(rv.w, wv.w, acc);
    }
    h[i] = acc;
}

__global__ void k_out_init(const float* __restrict__ h, const float* __restrict__ dis,
                           const float* __restrict__ b, float* __restrict__ out) {
    int i = blockIdx.x * blockDim.x + threadIdx.x;
    if (i >= NN) return;
    float d = dis[i];
    out[i] = fmaf(h[i], d * d, b[0]);
}

__global__ void k_out_edges(const float* __restrict__ h, const float* __restrict__ dis,
                            const int* __restrict__ src, const int* __restrict__ dst,
                            float* out) {
    int e = blockIdx.x * blockDim.x + threadIdx.x;
    if (e >= NE) return;
    int s = src[e], d = dst[e];
    atomic_add_f32(&out[d], h[s] * dis[s] * dis[d]);
}

__global__ void k_out_elu(float* x) {
    int i = blockIdx.x * blockDim.x + threadIdx.x;
    if (i < NN) x[i] = elu1(x[i]);
}

extern "C" void kernel_launch(void* const* d_in, const int* in_sizes, int n_in,
                              void* d_out, int out_size, void* d_ws, size_t ws_size,
                              hipStream_t stream) {
    const float* x    = (const float*)d_in[0];
    const int*   ei   = (const int*)d_in[1];
    const int*   src  = ei;
    const int*   dstv = ei + NE;
    const float* W1 = (const float*)d_in[2];  const float* b1 = (const float*)d_in[3];
    const float* W2 = (const float*)d_in[4];  const float* b2 = (const float*)d_in[5];
    const float* W3 = (const float*)d_in[6];  const float* b3 = (const float*)d_in[7];
    const float* W4 = (const float*)d_in[8];  const float* b4 = (const float*)d_in[9];
    const float* W5 = (const float*)d_in[10]; const float* b5 = (const float*)d_in[11];
    const float* g2 = (const float*)d_in[12]; const float* be2 = (const float*)d_in[13];
    const float* g3 = (const float*)d_in[14]; const float* be3 = (const float*)d_in[15];

    float* out = (float*)d_out;
    float* ws  = (float*)d_ws;
    float* dis = ws;                       // N floats
    float* red = ws + NN;                  // 2 floats (+pad to 16 for alignment)
    float* H   = ws + NN + 16;             // N*64 floats (linear output)
    float* AGG = H + (size_t)NN * 64;      // N*64 floats (aggregate / activation)

    const int B = 256;
    const int gN     = NN / B;             // 1024
    const int gE     = NE / B;             // 8192
    const int gNF64v = (NN * 16) / B;      // N*64/4 elems -> 16384
    const int gEF64v = (NE / B) * 16;      // E*64/4 threads -> 131072
    const int gNF16v = (NN * 4) / B;       // 4096
    const int gEF16v = (NE / B) * 4;       // 32768
    const int gWMMA  = NN / 16 / 4;        // 4096 blocks x 4 waves = N/16 tiles

    // degree -> dis = 1/sqrt(deg+1)
    k_init_deg<<<gN, B, 0, stream>>>(dis);
    k_count_deg<<<gE, B, 0, stream>>>(dis, dstv);
    k_rsqrt<<<gN, B, 0, stream>>>(dis);

    // ---- layer 1: 9 -> 64, LN(g2) + ELU ----
    k_linear9<<<(NN * 16) / B, B, 0, stream>>>(x, W1, H);
    k_agg_init4<<<gNF64v, B, 0, stream>>>(H, dis, b1, AGG, 6, red);
    k_agg_edges4<<<gEF64v, B, 0, stream>>>(H, dis, src, dstv, AGG, 6);
    k_stats<<<2048, B, 0, stream>>>((const float4*)AGG, (long)NN * 16, red);
    k_ln_elu4<<<gNF64v, B, 0, stream>>>(AGG, red, g2, be2, 6);

    // ---- layer 2: 64 -> 64 (WMMA), LN(g2) + ELU ----
    k_linear_wmma<<<gWMMA, 128, 0, stream>>>(AGG, W2, H, 64);
    k_agg_init4<<<gNF64v, B, 0, stream>>>(H, dis, b2, AGG, 6, red);
    k_agg_edges4<<<gEF64v, B, 0, stream>>>(H, dis, src, dstv, AGG, 6);
    k_stats<<<2048, B, 0, stream>>>((const float4*)AGG, (long)NN * 16, red);
    k_ln_elu4<<<gNF64v, B, 0, stream>>>(AGG, red, g2, be2, 6);

    // ---- layer 3: 64 -> 64 (WMMA), LN(g3) + ELU ----
    k_linear_wmma<<<gWMMA, 128, 0, stream>>>(AGG, W3, H, 64);
    k_agg_init4<<<gNF64v, B, 0, stream>>>(H, dis, b3, AGG, 6, red);
    k_agg_edges4<<<gEF64v, B, 0, stream>>>(H, dis, src, dstv, AGG, 6);
    k_stats<<<2048, B, 0, stream>>>((const float4*)AGG, (long)NN * 16, red);
    k_ln_elu4<<<gNF64v, B, 0, stream>>>(AGG, red, g3, be3, 6);

    // ---- layer 4: 64 -> 16 (WMMA), ELU ----
    k_linear_wmma<<<gWMMA, 128, 0, stream>>>(AGG, W4, H, 16);
    k_agg_init4<<<gNF16v, B, 0, stream>>>(H, dis, b4, AGG, 4, red);
    k_agg_edges4<<<gEF16v, B, 0, stream>>>(H, dis, src, dstv, AGG, 4);
    k_elu4<<<gNF16v, B, 0, stream>>>(AGG, (long)NN * 4);

    // ---- layer 5: 16 -> 1, ELU, straight into d_out ----
    k_linear16<<<gN, B, 0, stream>>>(AGG, W5, H);
    k_out_init<<<gN, B, 0, stream>>>(H, dis, b5, out);
    k_out_edges<<<gE, B, 0, stream>>>(H, dis, src, dstv, out);
    k_out_elu<<<gN, B, 0, stream>>>(out);
}